// Post_process_deconv_32744830665221
// MI455X (gfx1250) — compile-verified
//
#include <hip/hip_runtime.h>
#include <stdint.h>

#define Hc   512
#define Wc   512
#define Bc   8
#define TH   8                  // tile rows per workgroup
#define HALO 8                  // staged halo rows above/below tile
#define SROWS (TH + 2 * HALO)   // 24 rows -> 48 KB LDS
#define HWc  (Hc * Wc)

typedef float        v4f __attribute__((ext_vector_type(4)));
typedef unsigned int v4u __attribute__((ext_vector_type(4)));
typedef unsigned int v8u __attribute__((ext_vector_type(8)));

typedef const float __attribute__((address_space(1))) * gptr_t;

// Rare path: bilinear for 4 samples straight from the (L2-resident) global
// depth plane. Truly branchless: unconditional clamped loads + cndmask zeros.
// Correct for any sample position; noinline keeps the hot loop small.
__device__ __attribute__((noinline)) v4f slow_bilin4(
    const float* __restrict__ g, v4f py4, v4f px4) {
  gptr_t gg = (gptr_t)g;
  v4f r;
#pragma unroll
  for (int j = 0; j < 4; ++j) {
    float py = py4[j], px = px4[j];
    float fy = floorf(py), fx = floorf(px);
    float ty = py - fy, tx = px - fx;
    int yi = (int)fy, xi = (int)fx;
    int yc0 = min(max(yi, 0), Hc - 1), yc1 = min(max(yi + 1, 0), Hc - 1);
    int xc0 = min(max(xi, 0), Wc - 1), xc1 = min(max(xi + 1, 0), Wc - 1);
    // Unconditional safe loads (clamped), then mask.
    float r00 = gg[yc0 * Wc + xc0];
    float r01 = gg[yc0 * Wc + xc1];
    float r10 = gg[yc1 * Wc + xc0];
    float r11 = gg[yc1 * Wc + xc1];
    bool my0 = (unsigned)yi < (unsigned)Hc, my1 = (unsigned)(yi + 1) < (unsigned)Hc;
    bool mx0 = (unsigned)xi < (unsigned)Wc, mx1 = (unsigned)(xi + 1) < (unsigned)Wc;
    float v00 = (my0 && mx0) ? r00 : 0.0f;
    float v01 = (my0 && mx1) ? r01 : 0.0f;
    float v10 = (my1 && mx0) ? r10 : 0.0f;
    float v11 = (my1 && mx1) ? r11 : 0.0f;
    float top = v00 + tx * (v01 - v00);
    float bot = v10 + tx * (v11 - v10);
    r[j] = top + ty * (bot - top);
  }
  return r;
}

__global__ __launch_bounds__(256) void deform_post_kernel(
    const float* __restrict__ depth, const float* __restrict__ weight,
    const float* __restrict__ offset, float* __restrict__ out) {
  __shared__ float smem[SROWS * Wc];

  const int b   = blockIdx.y;
  const int y0  = blockIdx.x * TH;
  const int ylo = max(y0 - HALO, 0);
  const int yhi = min(y0 + TH - 1 + HALO, Hc - 1);
  const int nrows = yhi - ylo + 1;

  const float* dB = depth + (size_t)b * HWc;

  // ---- TDM: DMA depth rows [ylo, yhi] (full width) into LDS; wave 0 only ----
  if (threadIdx.x < 32) {
    uint64_t ga   = (uint64_t)(uintptr_t)(dB + (size_t)ylo * Wc);
    uint32_t lds0 = (uint32_t)(uintptr_t)&smem[0];

    v4u g0;
    g0[0] = 1u;                                                  // count=1 (valid), user mode
    g0[1] = lds0;                                                // lds_addr (bytes)
    g0[2] = (uint32_t)ga;                                        // global_addr[31:0]
    g0[3] = (uint32_t)((ga >> 32) & 0x01FFFFFFu) | (2u << 30);   // addr[56:32] | type=2

    v8u g1;
    g1[0] = 2u << 16;                                 // data_size = 4 bytes
    g1[1] = ((uint32_t)Wc & 0xFFFFu) << 16;           // tensor_dim0[15:0] = 512
    g1[2] = ((uint32_t)(Hc - ylo) & 0xFFFFu) << 16;   // tensor_dim1[15:0] | tdim0_hi=0
    g1[3] = ((uint32_t)Wc) << 16;                     // tile_dim0 = 512 | tdim1_hi=0
    g1[4] = (uint32_t)nrows;                          // tile_dim1 = staged rows
    g1[5] = (uint32_t)Wc;                             // tensor_dim0_stride = 512
    g1[6] = 0u;
    g1[7] = 0u;

    asm volatile("tensor_load_to_lds %0, %1" :: "s"(g0), "s"(g1) : "memory");
    __builtin_amdgcn_s_wait_tensorcnt(0);
  }
  __syncthreads();

  const float* wB = weight + (size_t)b * 9  * HWc;
  const float* oB = offset + (size_t)b * 18 * HWc;
  float*       oO = out    + (size_t)b * HWc;

#pragma unroll 1
  for (int it = 0; it < (TH * Wc) / (256 * 4); ++it) {
    const int idx = it * (256 * 4) + (int)threadIdx.x * 4;  // 4 consecutive columns
    const int x   = idx & (Wc - 1);
    const int y   = y0 + (idx >> 9);
    const size_t pix = (size_t)y * Wc + x;
    const float yf = (float)y;

    // Single-use weight stream: wide non-temporal loads.
    v4f wv[9];
    v4f msum = {0.f, 0.f, 0.f, 0.f};
#pragma unroll
    for (int k = 0; k < 9; ++k) {
      wv[k] = __builtin_nontemporal_load((const v4f*)(wB + (size_t)k * HWc + pix));
      msum += wv[k];
    }
    const v4f mean = msum * (1.0f / 9.0f);

    // acc = sum_k s_k * w_k ; ssum = sum_k s_k ; result = acc - mean*ssum + d
    v4f acc  = {0.f, 0.f, 0.f, 0.f};
    v4f ssum = {0.f, 0.f, 0.f, 0.f};
#pragma unroll
    for (int k = 0; k < 9; ++k) {
      v4f dyv = __builtin_nontemporal_load((const v4f*)(oB + (size_t)(2 * k)     * HWc + pix));
      v4f dxv = __builtin_nontemporal_load((const v4f*)(oB + (size_t)(2 * k + 1) * HWc + pix));
      const float kyf = (float)(k / 3 - 1);
      const float kxf = (float)(k % 3 - 1);

      // Compute all 4 sample positions, then ONE fast/slow branch per tap-group.
      v4f py4, px4;
      bool fast = true;
#pragma unroll
      for (int j = 0; j < 4; ++j) {
        const float py = dyv[j] + yf + kyf;
        const float px = dxv[j] + (float)(x + j) + kxf;
        py4[j] = py;
        px4[j] = px;
        const int yi = (int)floorf(py);
        fast = fast && (yi >= ylo) && (yi < yhi);   // both rows in staged window
      }

      v4f sv4;
      if (__builtin_expect(fast, 1)) {
        // Branchless LDS gathers; only x-bound cndmask selects.
#pragma unroll
        for (int j = 0; j < 4; ++j) {
          const float py = py4[j], px = px4[j];
          const float fy = floorf(py), fx = floorf(px);
          const float ty = py - fy,    tx = px - fx;
          const int yi = (int)fy, xi = (int)fx;
          const int yw = yi - ylo;
          const int xc0 = min(max(xi, 0), Wc - 1);
          const int xc1 = min(max(xi + 1, 0), Wc - 1);
          const float* row0 = smem + yw * Wc;
          const float l00 = row0[xc0],      l01 = row0[xc1];
          const float l10 = row0[Wc + xc0], l11 = row0[Wc + xc1];
          const bool mx0 = (unsigned)xi       < (unsigned)Wc;
          const bool mx1 = (unsigned)(xi + 1) < (unsigned)Wc;
          const float v00 = mx0 ? l00 : 0.0f, v10 = mx0 ? l10 : 0.0f;
          const float v01 = mx1 ? l01 : 0.0f, v11 = mx1 ? l11 : 0.0f;
          const float top = v00 + tx * (v01 - v00);
          const float bot = v10 + tx * (v11 - v10);
          sv4[j] = top + ty * (bot - top);
        }
      } else {
        sv4 = slow_bilin4(dB, py4, px4);
      }

#pragma unroll
      for (int j = 0; j < 4; ++j) {
        acc[j]  = fmaf(sv4[j], wv[k][j], acc[j]);
        ssum[j] += sv4[j];
      }
    }

    // Center depth straight from the staged LDS window (b128 LDS read).
    const v4f dc  = *(const v4f*)(&smem[(y - ylo) * Wc + x]);
    const v4f res = acc - mean * ssum + dc;
    __builtin_nontemporal_store(res, (v4f*)(oO + pix));
  }
}

extern "C" void kernel_launch(void* const* d_in, const int* in_sizes, int n_in,
                              void* d_out, int out_size, void* d_ws, size_t ws_size,
                              hipStream_t stream) {
  (void)in_sizes; (void)n_in; (void)d_ws; (void)ws_size; (void)out_size;
  const float* depth  = (const float*)d_in[0];  // (8,1,512,512)  f32
  const float* weight = (const float*)d_in[1];  // (8,9,512,512)  f32
  const float* offset = (const float*)d_in[2];  // (8,18,512,512) f32
  float* out = (float*)d_out;                   // (8,1,512,512)  f32

  dim3 grid(Hc / TH, Bc);
  deform_post_kernel<<<grid, 256, 0, stream>>>(depth, weight, offset, out);
}